// GotenInteractionModule_14791867367987
// MI455X (gfx1250) — compile-verified
//
#include <hip/hip_runtime.h>
#include <hip/hip_bf16.h>
#include <math.h>

typedef float v2f __attribute__((ext_vector_type(2)));
typedef float v8f __attribute__((ext_vector_type(8)));

#define GEMM_BLOCK 256

// ---------------------------------------------------------------------------
// Generic fp32 WMMA GEMM: C[M,N] = (A[M,K] @ B[K,N]) * scale
// Requirements: K % 4 == 0 (true for every call site). M, N arbitrary.
// Block: 8 waves = 4 row-tiles x 2 col-tiles -> 64x64 of C per block.
// Each wave: 16x32 of C via two accumulator chains (hides WMMA RAW latency).
// A staged row-major in LDS (fragment = contiguous 8B pair -> ds_load_b64).
// B staged k-pair-interleaved in LDS: BsP[p][c] = {B[2p][c], B[2p+1][c]} so a
// lane's B fragment is one aligned 8B read; pair-row stride 160 (mod 64 = 32)
// keeps the two half-waves on disjoint bank halves. Columns beyond N are
// zero-filled at stage time so EXEC stays all-ones around v_wmma.
// ---------------------------------------------------------------------------
__global__ __launch_bounds__(GEMM_BLOCK)
void wmma_gemm_f32(const float* __restrict__ A, const float* __restrict__ B,
                   float* __restrict__ C, int M, int N, int K,
                   int lda, int ldb, int ldc, float scale)
{
    __shared__ float As[64 * 68];        // 64 rows x kc (<=64), row stride 68
    __shared__ float Bs[32 * 160];       // 32 k-pairs x 64 cols x 2, stride 160
    const int tid  = threadIdx.x;
    const int wave = tid >> 5;
    const int lane = tid & 31;
    const int r    = lane & 15;
    const int half = lane >> 4;

    const int rowTile = wave >> 1;                 // 0..3
    const int colTile = wave & 1;                  // 0..1
    const int rowBase = blockIdx.x * 64;
    const int row0    = rowBase + rowTile * 16;
    const int colBase = blockIdx.y * 64;
    const int c0      = colTile * 32;              // local col of acc0

    v8f acc0 = {}, acc1 = {};
    for (int k0 = 0; k0 < K; k0 += 64) {
        const int kc = (K - k0) < 64 ? (K - k0) : 64;   // multiple of 4
        __syncthreads();
        // stage A chunk: 64 rows x kc (row-clamped for M % 64 != 0)
        for (int idx = tid; idx < 64 * kc; idx += GEMM_BLOCK) {
            int rr = idx / kc;
            int kk = idx - rr * kc;
            int gr = rowBase + rr;
            gr = gr < M ? gr : M - 1;
            As[rr * 68 + kk] = A[(long)gr * lda + k0 + kk];
        }
        // stage B chunk as k-pairs: (kc/2) pairs x 64 cols (zero-fill beyond N)
        for (int idx = tid; idx < (kc >> 1) * 64; idx += GEMM_BLOCK) {
            int p  = idx >> 6;
            int cl = idx & 63;
            int gc = colBase + cl;
            float b0 = 0.0f, b1 = 0.0f;
            if (gc < N) {
                long col = (long)gc;
                b0 = B[(long)(k0 + 2 * p)     * ldb + col];
                b1 = B[(long)(k0 + 2 * p + 1) * ldb + col];
            }
            Bs[p * 160 + 2 * cl]     = b0;
            Bs[p * 160 + 2 * cl + 1] = b1;
        }
        __syncthreads();
        const float* as = As + (rowTile * 16 + r) * 68;
        #pragma unroll 4
        for (int kk = 0; kk < kc; kk += 4) {
            v2f a, b0, b1;
            a.x = as[kk + 2 * half];
            a.y = as[kk + 2 * half + 1];
            const float* bp = Bs + ((kk >> 1) + half) * 160;
            b0 = *(const v2f*)(bp + 2 * (c0 + r));
            b1 = *(const v2f*)(bp + 2 * (c0 + 16 + r));
            acc0 = __builtin_amdgcn_wmma_f32_16x16x4_f32(
                       false, a, false, b0, (short)0, acc0, false, false);
            acc1 = __builtin_amdgcn_wmma_f32_16x16x4_f32(
                       false, a, false, b1, (short)0, acc1, false, false);
        }
    }
    const int colA = colBase + c0 + r;
    const int colB = colA + 16;
    #pragma unroll
    for (int j = 0; j < 8; ++j) {
        int row = row0 + j + 8 * half;
        if (row < M) {
            if (colA < N) C[(long)row * ldc + colA] = acc0[j] * scale;
            if (colB < N) C[(long)row * ldc + colB] = acc1[j] * scale;
        }
    }
}

// ---------------------------------------------------------------------------
// Wave-per-row LayerNorm over 64 features. res may be null. In-place safe.
// ---------------------------------------------------------------------------
__global__ void ln64_kernel(const float* __restrict__ x, const float* __restrict__ res,
                            const float* __restrict__ g, const float* __restrict__ b,
                            float* __restrict__ out, int rows)
{
    int row = blockIdx.x * 8 + (threadIdx.x >> 5);
    if (row >= rows) return;
    int lane = threadIdx.x & 31;
    long base = (long)row * 64;
    float v0 = x[base + lane];
    float v1 = x[base + lane + 32];
    if (res) { v0 += res[base + lane]; v1 += res[base + lane + 32]; }
    float s = v0 + v1, s2 = v0 * v0 + v1 * v1;
    #pragma unroll
    for (int o = 16; o > 0; o >>= 1) {
        s  += __shfl_xor(s,  o, 32);
        s2 += __shfl_xor(s2, o, 32);
    }
    float mu  = s * (1.0f / 64.0f);
    float var = s2 * (1.0f / 64.0f) - mu * mu;
    float inv = rsqrtf(var + 1e-5f);
    out[base + lane]      = (v0 - mu) * inv * g[lane]      + b[lane];
    out[base + lane + 32] = (v1 - mu) * inv * g[lane + 32] + b[lane + 32];
}

// pe = pn[en] * ew ; scatter-add into cat[:, 64:128]
__global__ void scatter_pe_kernel(const float* __restrict__ pn, const float* __restrict__ ew,
                                  const int* __restrict__ en, const int* __restrict__ ec,
                                  float* __restrict__ cat, int total)
{
    int i = blockIdx.x * 256 + threadIdx.x;
    if (i >= total) return;
    int e = i >> 6, c = i & 63;
    float v = pn[(long)en[e] * 64 + c] * ew[(long)e * 64 + c];
    atomicAdd(&cat[(long)ec[e] * 128 + 64 + c], v);
}

// t_cat = [h[ec] | h[en] | phi]
__global__ void build_tcat_kernel(const float* __restrict__ h, const float* __restrict__ phi,
                                  const int* __restrict__ ec, const int* __restrict__ en,
                                  float* __restrict__ tcat, int total)
{
    int i = blockIdx.x * 256 + threadIdx.x;
    if (i >= total) return;
    int e = i / 136, c = i - e * 136;
    float v;
    if (c < 64)       v = h[(long)ec[e] * 64 + c];
    else if (c < 128) v = h[(long)en[e] * 64 + (c - 64)];
    else              v = phi[(long)e * 8 + (c - 128)];
    tcat[i] = v;
}

__global__ void gather_hj_kernel(const float* __restrict__ h, const int* __restrict__ en,
                                 float* __restrict__ hj, int total)
{
    int i = blockIdx.x * 256 + threadIdx.x;
    if (i >= total) return;
    int e = i >> 6, c = i & 63;
    hj[i] = h[(long)en[e] * 64 + c];
}

__global__ void mul_kernel(float* __restrict__ a, const float* __restrict__ b, int total)
{
    int i = blockIdx.x * 256 + threadIdx.x;
    if (i >= total) return;
    a[i] *= b[i];
}

__device__ __forceinline__ int irrep_of(int d) { return (d == 0) ? 0 : ((d < 4) ? 1 : 2); }

// env-weighter + segment-sum: Xacc[seg][m][d] += sph[e][d] * w0[e][IDX[d]*8+m]
__global__ void envw_scatter_kernel(const float* __restrict__ sph, const float* __restrict__ w0,
                                    const int* __restrict__ ec, float* __restrict__ Xacc, int total)
{
    int i = blockIdx.x * 256 + threadIdx.x;
    if (i >= total) return;
    int e = i / 72, md = i - e * 72;
    int m = md / 9, d = md - m * 9;
    int k = irrep_of(d);
    float v = sph[(long)e * 9 + d] * w0[(long)e * 24 + k * 8 + m];
    atomicAdd(&Xacc[(long)ec[e] * 72 + m * 9 + d], v);
}

// SO(3) norm per node (optionally with additive term). In-place safe.
__global__ void so3_kernel(const float* __restrict__ Xin, const float* __restrict__ Xadd,
                           float* __restrict__ Xout, int nodes)
{
    int n = blockIdx.x * 256 + threadIdx.x;
    if (n >= nodes) return;
    float buf[72];
    long base = (long)n * 72;
    for (int t = 0; t < 72; ++t) {
        float v = Xin[base + t];
        if (Xadd) v += Xadd[base + t];
        buf[t] = v;
    }
    const int off[3] = {0, 1, 4};
    const int dd[3]  = {1, 3, 5};
    for (int k = 0; k < 3; ++k) {
        float n2 = 0.0f;
        for (int m = 0; m < 8; ++m)
            for (int j = 0; j < dd[k]; ++j) {
                float v = buf[m * 9 + off[k] + j];
                n2 += v * v;
            }
        n2 *= (1.0f / 8.0f);
        float inv = rsqrtf(n2 + 1e-5f);
        for (int m = 0; m < 8; ++m)
            for (int j = 0; j < dd[k]; ++j)
                Xout[base + m * 9 + off[k] + j] = buf[m * 9 + off[k] + j] * inv;
    }
}

// hacc[seg][c] += env[e][c] (c < 64, env row stride 280)
__global__ void scatter_cols64_kernel(const float* __restrict__ env, const int* __restrict__ ec,
                                      float* __restrict__ hacc, int total)
{
    int i = blockIdx.x * 256 + threadIdx.x;
    if (i >= total) return;
    int e = i >> 6, c = i & 63;
    atomicAdd(&hacc[(long)ec[e] * 64 + c], env[(long)e * 280 + c]);
}

__device__ __forceinline__ unsigned enc_ord(float f)
{
    unsigned u = __float_as_uint(f);
    return (u & 0x80000000u) ? ~u : (u | 0x80000000u);
}
__device__ __forceinline__ float dec_ord(unsigned u)
{
    u = (u & 0x80000000u) ? (u & 0x7FFFFFFFu) : ~u;
    return __uint_as_float(u);
}

// att[e][m] = 4 * sum_h Q[e][m][h]*K[e][m][h]; atomicMax into per-segment max
__global__ void att_max_kernel(const float* __restrict__ qk, const int* __restrict__ ec,
                               float* __restrict__ att, unsigned* __restrict__ nmax, int total)
{
    int i = blockIdx.x * 256 + threadIdx.x;
    if (i >= total) return;
    int e = i >> 3, m = i & 7;
    const float* q = qk + (long)e * 256 + m * 16;
    const float* k = qk + (long)e * 256 + 128 + m * 16;
    float s = 0.0f;
    #pragma unroll
    for (int h = 0; h < 16; ++h) s += q[h] * k[h];
    s *= 4.0f;
    att[i] = s;
    atomicMax(&nmax[(long)ec[e] * 8 + m], enc_ord(s));
}

__global__ void exp_sum_kernel(const float* __restrict__ att, const unsigned* __restrict__ nmax,
                               const int* __restrict__ ec, float* __restrict__ expv,
                               float* __restrict__ nsum, int total)
{
    int i = blockIdx.x * 256 + threadIdx.x;
    if (i >= total) return;
    int e = i >> 3, m = i & 7;
    long seg = (long)ec[e] * 8 + m;
    float ev = __expf(att[i] - dec_ord(nmax[seg]));
    expv[i] = ev;
    atomicAdd(&nsum[seg], ev);
}

// x = d_sph + d_eq ; Xacc[seg] += x * a
__global__ void deq_scatter_kernel(const float* __restrict__ sph, const float* __restrict__ env,
                                   const float* __restrict__ X, const float* __restrict__ expv,
                                   const float* __restrict__ nsum, const int* __restrict__ ec,
                                   const int* __restrict__ en, float* __restrict__ Xacc, int total)
{
    int i = blockIdx.x * 256 + threadIdx.x;
    if (i >= total) return;
    int e = i / 72, md = i - e * 72;
    int m = md / 9, d = md - m * 9;
    int k = irrep_of(d);
    int seg = ec[e];
    float a = expv[(long)e * 8 + m] / (nsum[(long)seg * 8 + m] + 1e-12f);
    const float* er = env + (long)e * 280;
    float dsph = sph[(long)e * 9 + d] * er[k * 8 + m];
    const float* wl = er + 24 + k * 64;        // [i*8 + m]
    const float* xr = X + (long)en[e] * 72;    // [i*9 + d]
    float deq = 0.0f;
    #pragma unroll
    for (int ii = 0; ii < 8; ++ii) deq += xr[ii * 9 + d] * wl[ii * 8 + m];
    float val = dsph + deq * 0.35355339f;
    atomicAdd(&Xacc[(long)seg * 72 + m * 9 + d], val * a);
}

// final equivariant linear + node_out layout
__global__ void eq_out_kernel(const float* __restrict__ X, const float* __restrict__ wbuf,
                              float* __restrict__ out, int total)
{
    int i = blockIdx.x * 256 + threadIdx.x;
    if (i >= total) return;
    int n = i / 576, rem = i - n * 576;
    int k, oc, j;
    if (rem < 64)       { k = 0; oc = rem;             j = 0;                 }
    else if (rem < 256) { k = 1; oc = (rem - 64) / 3;  j = (rem - 64) % 3;    }
    else                { k = 2; oc = (rem - 256) / 5; j = (rem - 256) % 5;   }
    const int off[3] = {0, 1, 4};
    int d = off[k] + j;
    const float* w  = wbuf + (long)n * 1536 + k * 512;  // [i*64 + oc]
    const float* xr = X + (long)n * 72;                 // [i*9 + d]
    float s = 0.0f;
    #pragma unroll
    for (int ii = 0; ii < 8; ++ii) s += xr[ii * 9 + d] * w[ii * 64 + oc];
    out[i] = s * 0.35355339f;
}

// ---------------------------------------------------------------------------

static inline int cdiv(int a, int b) { return (a + b - 1) / b; }

extern "C" void kernel_launch(void* const* d_in, const int* in_sizes, int n_in,
                              void* d_out, int out_size, void* d_ws, size_t ws_size,
                              hipStream_t stream)
{
    (void)n_in; (void)out_size; (void)ws_size;

    const float* node_attr  = (const float*)d_in[0];
    const float* phi        = (const float*)d_in[1];
    const float* sph        = (const float*)d_in[2];
    /* edge_length d_in[3] unused by reference */
    const float* W_node     = (const float*)d_in[4];
    const float* W_cnode    = (const float*)d_in[5];
    const float* W_concat   = (const float*)d_in[6];
    const float* W_edge     = (const float*)d_in[7];
    const float* ln0_g      = (const float*)d_in[8];
    const float* ln0_b      = (const float*)d_in[9];
    const float* lne_g      = (const float*)d_in[10];
    const float* lne_b      = (const float*)d_in[11];
    const float* W_tcat     = (const float*)d_in[12];
    const float* W_temb0    = (const float*)d_in[13];
    const float* W_hjemb0   = (const float*)d_in[14];
    const float* W_rs       = (const float*)d_in[15];
    const float* W_hj       = (const float*)d_in[16];
    const float* lnl_g      = (const float*)d_in[17];
    const float* lnl_b      = (const float*)d_in[18];
    const float* W_q        = (const float*)d_in[19];
    const float* W_k        = (const float*)d_in[20];
    const float* W_hemb     = (const float*)d_in[21];
    const float* W_tij_emb  = (const float*)d_in[22];
    const int*   ec         = (const int*)d_in[23];
    const int*   en         = (const int*)d_in[24];

    const int Nn = in_sizes[0] / 32;   // 10000
    const int E  = in_sizes[1] / 8;    // 320000

    float* out = (float*)d_out;

    // -------- workspace carve-up (floats) --------
    float* ws = (float*)d_ws;
    size_t o = 0;
    float* pn    = ws + o; o += (size_t)Nn * 64;
    float* cat   = ws + o; o += (size_t)Nn * 128;
    float* h     = ws + o; o += (size_t)Nn * 64;
    float* hacc  = ws + o; o += (size_t)Nn * 64;     // also pre-LN h buffer
    float* X     = ws + o; o += (size_t)Nn * 72;
    float* Xacc  = ws + o; o += (size_t)Nn * 72;
    unsigned* nmax = (unsigned*)(ws + o); o += (size_t)Nn * 8;
    float* nsum  = ws + o; o += (size_t)Nn * 8;
    float* wbuf  = ws + o; o += (size_t)Nn * 1536;
    float* tij   = ws + o; o += (size_t)E * 64;
    float* hj    = ws + o; o += (size_t)E * 64;
    float* tcat  = ws + o; o += (size_t)E * 136;     // reused: ew, then att/expv
    float* eb0   = ws + o; o += (size_t)E * 280;     // env / w0 / pre-LN t_ij
    float* eb1   = ws + o; o += (size_t)E * 280;     // env rhs / QK
    float* ew    = tcat;
    float* att   = tcat;
    float* expv  = tcat + (size_t)E * 8;

    const float sc64   = 0.125f;                 // 64^-0.5
    const float sc136  = 1.0f / sqrtf(136.0f);   // (2*LAT+D_RAD)^-0.5

    #define GEMM(A, B, C, M, N, K, lda, ldb, ldc, sc) \
        wmma_gemm_f32<<<dim3(cdiv((M), 64), cdiv((N), 64)), GEMM_BLOCK, 0, stream>>>( \
            (A), (B), (C), (M), (N), (K), (lda), (ldb), (ldc), (sc))
    #define EW_LAUNCH(total) <<<cdiv((total), 256), 256, 0, stream>>>

    // ---- node embedding + message ----
    hipMemsetAsync(cat, 0, (size_t)Nn * 128 * sizeof(float), stream);
    GEMM(node_attr, W_node,  pn,  Nn, 64, 32, 32, 64, 64,  1.0f);
    GEMM(node_attr, W_cnode, cat, Nn, 64, 32, 32, 64, 128, 1.0f);   // cat[:, :64] = pc
    GEMM(phi,       W_edge,  ew,  E,  64, 8,  8,  64, 64,  1.0f);
    scatter_pe_kernel EW_LAUNCH(E * 64)(pn, ew, en, ec, cat, E * 64);
    GEMM(cat, W_concat, hacc, Nn, 64, 128, 128, 64, 64, 1.0f);
    ln64_kernel<<<cdiv(Nn, 8), 256, 0, stream>>>(hacc, nullptr, ln0_g, ln0_b, h, Nn);

    // ---- t_ij ----
    build_tcat_kernel EW_LAUNCH(E * 136)(h, phi, ec, en, tcat, E * 136);
    GEMM(tcat, W_tcat, eb0, E, 64, 136, 136, 64, 64, sc136);
    ln64_kernel<<<cdiv(E, 8), 256, 0, stream>>>(eb0, nullptr, lne_g, lne_b, tij, E);

    // ---- initial environment X ----
    gather_hj_kernel EW_LAUNCH(E * 64)(h, en, hj, E * 64);
    GEMM(tij, W_temb0,  eb0, E, 24, 64, 64, 24, 24, sc64);
    GEMM(hj,  W_hjemb0, eb1, E, 24, 64, 64, 24, 24, sc64);
    mul_kernel EW_LAUNCH(E * 24)(eb0, eb1, E * 24);
    hipMemsetAsync(Xacc, 0, (size_t)Nn * 72 * sizeof(float), stream);
    envw_scatter_kernel EW_LAUNCH(E * 72)(sph, eb0, ec, Xacc, E * 72);
    so3_kernel EW_LAUNCH(Nn)(Xacc, nullptr, X, Nn);

    // ---- interaction layers ----
    for (int l = 0; l < 2; ++l) {
        gather_hj_kernel EW_LAUNCH(E * 64)(h, en, hj, E * 64);   // h_j (pre-update)

        GEMM(tij, W_rs + (size_t)l * 64 * 280, eb0, E, 280, 64, 64, 280, 280, 1.0f);
        GEMM(hj,  W_hj + (size_t)l * 64 * 280, eb1, E, 280, 64, 64, 280, 280, sc64);
        mul_kernel EW_LAUNCH(E * 280)(eb0, eb1, E * 280);        // env

        hipMemsetAsync(hacc, 0, (size_t)Nn * 64 * sizeof(float), stream);
        scatter_cols64_kernel EW_LAUNCH(E * 64)(eb0, ec, hacc, E * 64);
        ln64_kernel<<<cdiv(Nn, 8), 256, 0, stream>>>(h, hacc, lnl_g + l * 64, lnl_b + l * 64, h, Nn);

        // attention (Q from t_ij, K from old h_j)
        GEMM(tij, W_q + (size_t)l * 64 * 128, eb1,       E, 128, 64, 64, 128, 256, 1.0f);
        GEMM(hj,  W_k + (size_t)l * 64 * 128, eb1 + 128, E, 128, 64, 64, 128, 256, 1.0f);
        hipMemsetAsync(nmax, 0, (size_t)Nn * 8 * sizeof(unsigned), stream);
        hipMemsetAsync(nsum, 0, (size_t)Nn * 8 * sizeof(float), stream);
        att_max_kernel EW_LAUNCH(E * 8)(eb1, ec, att, nmax, E * 8);
        exp_sum_kernel EW_LAUNCH(E * 8)(att, nmax, ec, expv, nsum, E * 8);

        hipMemsetAsync(Xacc, 0, (size_t)Nn * 72 * sizeof(float), stream);
        deq_scatter_kernel EW_LAUNCH(E * 72)(sph, eb0, X, expv, nsum, ec, en, Xacc, E * 72);
        so3_kernel EW_LAUNCH(Nn)(X, Xacc, X, Nn);
    }

    // ---- outputs ----
    GEMM(h, W_hemb, wbuf, Nn, 1536, 64, 64, 1536, 1536, sc64);
    eq_out_kernel EW_LAUNCH(Nn * 576)(X, wbuf, out, Nn * 576);
    GEMM(tij, W_tij_emb, out + (size_t)Nn * 576, E, 64, 64, 64, 64, 64, sc64);

    #undef GEMM
    #undef EW_LAUNCH
}